// KnnSim_1443109012314
// MI455X (gfx1250) — compile-verified
//
#include <hip/hip_runtime.h>

typedef float v2f __attribute__((ext_vector_type(2)));
typedef float v8f __attribute__((ext_vector_type(8)));

#define B_      2048
#define N_      50000
#define D_      256
#define TOPK    50
#define NSLICES 8
#define SLICE   (N_ / NSLICES)   // 6250
#define SUB     512              // sub-chunk columns held in LDS
#define RUNOFF  SUB              // running top-k list lives at cols [SUB, SUB+TOPK)
#define SIMW    (SUB + TOPK + 2) // 564: pad so lanes 0..31 hit distinct banks
#define MROWS   32               // batch rows per block (M blocking for B reuse)
#define NEGINF  (-__builtin_inff())

// ---------------------------------------------------------------------------
// Kernel 1: fused fp32 WMMA GEMM (16x16x4) + per-slice streaming top-50.
// Block = 32 batch rows x one N-slice. 16 waves; wave w owns N-tiles {w, w+16}
// and both M-halves: 4 WMMAs per K-step sharing 2 A-frags + 2 B-frags
// => each B byte from L2 feeds 2 WMMAs (~16 FLOP/B of L2 traffic).
// LDS ~124 KB (CDNA5 allows up to 320 KB per workgroup).
// ---------------------------------------------------------------------------
__global__ __launch_bounds__(512)
void knn_gemm_topk(const float* __restrict__ feat,
                   const float* __restrict__ anch,
                   float* __restrict__ wsval,
                   int*   __restrict__ wsidx)
{
    __shared__ float Als[MROWS][257];      // A tile, stride 257 => conflict-free
    __shared__ float sim[MROWS][SIMW];     // sub-chunk sims + running top-k vals
    __shared__ int   runidx[MROWS][TOPK];  // global anchor idx of running list
    __shared__ float stgv[MROWS][TOPK];    // extraction staging
    __shared__ int   stgi[MROWS][TOPK];

    const int tid     = threadIdx.x;
    const int lane    = tid & 31;
    const int wave    = tid >> 5;          // 0..15
    const int slice   = blockIdx.x;
    const int rowbase = blockIdx.y * MROWS;

    // Load A tile (coalesced) and init running list to -inf.
    for (int i = tid; i < MROWS * D_; i += 512) {
        int r = i >> 8, d = i & 255;
        Als[r][d] = feat[(size_t)(rowbase + r) * D_ + d];
    }
    for (int i = tid; i < MROWS * (TOPK + 2); i += 512) {
        int r = i / (TOPK + 2), j = i % (TOPK + 2);
        sim[r][RUNOFF + j] = NEGINF;
        if (j < TOPK) runidx[r][j] = 0;
    }
    __syncthreads();

    const int m    = lane & 15;            // M (for A) / N (for B) position
    const int koff = (lane < 16) ? 0 : 2;  // K sub-position per ISA f32 layout

    for (int cb = 0; cb < SLICE; cb += SUB) {
        const int nbase = slice * SLICE + cb;
        const int valid = (SLICE - cb < SUB) ? (SLICE - cb) : SUB;

        // ---- compute: wave w handles N-tiles t0=w, t1=w+16, both M-halves.
        {
            const int t0 = wave;
            const int t1 = wave + 16;
            int n0 = nbase + t0 * 16 + m; n0 = (n0 < N_ - 1) ? n0 : (N_ - 1);
            int n1 = nbase + t1 * 16 + m; n1 = (n1 < N_ - 1) ? n1 : (N_ - 1);
            const float* bp0 = anch + (size_t)n0 * D_ + koff;
            const float* bp1 = anch + (size_t)n1 * D_ + koff;
            v8f c00 = {0.f,0.f,0.f,0.f,0.f,0.f,0.f,0.f};   // rows 0-15  x t0
            v8f c10 = c00;                                  // rows 16-31 x t0
            v8f c01 = c00;                                  // rows 0-15  x t1
            v8f c11 = c00;                                  // rows 16-31 x t1
            #pragma unroll 4
            for (int k = 0; k < D_; k += 4) {
                v2f a0; a0.x = Als[m][k + koff];      a0.y = Als[m][k + koff + 1];
                v2f a1; a1.x = Als[m + 16][k + koff]; a1.y = Als[m + 16][k + koff + 1];
                v2f b0; b0.x = bp0[k]; b0.y = bp0[k + 1];
                v2f b1; b1.x = bp1[k]; b1.y = bp1[k + 1];
                c00 = __builtin_amdgcn_wmma_f32_16x16x4_f32(false, a0, false, b0,
                                                            (short)0, c00, false, false);
                c10 = __builtin_amdgcn_wmma_f32_16x16x4_f32(false, a1, false, b0,
                                                            (short)0, c10, false, false);
                c01 = __builtin_amdgcn_wmma_f32_16x16x4_f32(false, a0, false, b1,
                                                            (short)0, c01, false, false);
                c11 = __builtin_amdgcn_wmma_f32_16x16x4_f32(false, a1, false, b1,
                                                            (short)0, c11, false, false);
            }
            const int mrow = (lane < 16) ? 0 : 8;
            const int col0 = t0 * 16 + m;
            const int col1 = t1 * 16 + m;
            const bool v0 = (col0 < valid);
            const bool v1 = (col1 < valid);
            #pragma unroll
            for (int r = 0; r < 8; ++r) {
                sim[mrow + r][col0]      = v0 ? c00[r] : NEGINF;
                sim[16 + mrow + r][col0] = v0 ? c10[r] : NEGINF;
                sim[mrow + r][col1]      = v1 ? c01[r] : NEGINF;
                sim[16 + mrow + r][col1] = v1 ? c11[r] : NEGINF;
            }
        }
        __syncthreads();

        // ---- selection: wave w extracts top-50 for rows 2w, 2w+1 over
        //      (sub-chunk sims ++ running list); this is also the merge step.
        for (int rr = 0; rr < 2; ++rr) {
            const int row = wave * 2 + rr;
            for (int tsel = 0; tsel < TOPK; ++tsel) {
                float bv = NEGINF; int bg = 0x7fffffff; int bc = 0;
                for (int cc = lane; cc < RUNOFF + TOPK; cc += 32) {
                    float v = sim[row][cc];
                    int   g = (cc < RUNOFF) ? (nbase + cc) : runidx[row][cc - RUNOFF];
                    if (v > bv || (v == bv && g < bg)) { bv = v; bg = g; bc = cc; }
                }
                #pragma unroll
                for (int off = 16; off; off >>= 1) {
                    float ov = __shfl_xor(bv, off, 32);
                    int   og = __shfl_xor(bg, off, 32);
                    int   oc = __shfl_xor(bc, off, 32);
                    if (ov > bv || (ov == bv && og < bg)) { bv = ov; bg = og; bc = oc; }
                }
                if (lane == 0) {
                    stgv[row][tsel] = bv;
                    stgi[row][tsel] = bg;
                    sim[row][bc]    = NEGINF;   // same-wave LDS ops stay ordered
                }
            }
            for (int j = lane; j < TOPK; j += 32) {   // commit new running list
                sim[row][RUNOFF + j] = stgv[row][j];
                runidx[row][j]       = stgi[row][j];
            }
        }
        __syncthreads();
    }

    // ---- emit sorted per-slice top-50 candidates to workspace
    for (int i = tid; i < MROWS * TOPK; i += 512) {
        int r = i / TOPK, j = i % TOPK;
        size_t o = ((size_t)(rowbase + r) * NSLICES + slice) * TOPK + j;
        wsval[o] = sim[r][RUNOFF + j];
        wsidx[o] = runidx[r][j];
    }
}

// ---------------------------------------------------------------------------
// Kernel 2: one wave per batch row; merge 8x50 candidates -> global top-50,
// gather anchor labels, emit loss and mean_sim.
// ---------------------------------------------------------------------------
__global__ __launch_bounds__(32)
void knn_merge(const float* __restrict__ wsval,
               const int*   __restrict__ wsidx,
               const int*   __restrict__ labels,
               const int*   __restrict__ albl,
               float*       __restrict__ out)
{
    __shared__ float cv[NSLICES * TOPK];
    __shared__ int   ci[NSLICES * TOPK];
    const int row  = blockIdx.x;
    const int lane = threadIdx.x;

    for (int i = lane; i < NSLICES * TOPK; i += 32) {
        size_t o = (size_t)row * NSLICES * TOPK + i;
        cv[i] = wsval[o];
        ci[i] = wsidx[o];
    }
    __syncthreads();

    const int myl = labels[row];
    float sum = 0.f; int match = 0;

    for (int t = 0; t < TOPK; ++t) {
        float bv = NEGINF; int bg = 0x7fffffff; int bc = 0;
        for (int i = lane; i < NSLICES * TOPK; i += 32) {
            float v = cv[i]; int g = ci[i];
            if (v > bv || (v == bv && g < bg)) { bv = v; bg = g; bc = i; }
        }
        #pragma unroll
        for (int off = 16; off; off >>= 1) {
            float ov = __shfl_xor(bv, off, 32);
            int   og = __shfl_xor(bg, off, 32);
            int   oc = __shfl_xor(bc, off, 32);
            if (ov > bv || (ov == bv && og < bg)) { bv = ov; bg = og; bc = oc; }
        }
        sum += bv;                          // broadcast-identical in all lanes
        match += (albl[bg] == myl) ? 1 : 0;
        if (lane == 0) cv[bc] = NEGINF;
        __syncthreads();                    // single-wave block: cheap
    }

    if (lane == 0) {
        out[row]      = -((float)match / (float)TOPK); // loss (NaN scrub is a no-op)
        out[B_ + row] = sum / (float)TOPK;             // mean_sim
    }
}

// ---------------------------------------------------------------------------
extern "C" void kernel_launch(void* const* d_in, const int* in_sizes, int n_in,
                              void* d_out, int out_size, void* d_ws, size_t ws_size,
                              hipStream_t stream) {
    const float* feat   = (const float*)d_in[0];   // [2048,256]
    const float* anch   = (const float*)d_in[1];   // [50000,256]
    const int*   labels = (const int*)d_in[2];     // [2048]
    /* d_in[3] = t_labels, unused */
    const int*   albl   = (const int*)d_in[4];     // [50000]
    float* out = (float*)d_out;                    // [2*2048]: loss ++ mean_sim

    float* wsval = (float*)d_ws;
    int*   wsidx = (int*)((char*)d_ws +
                          (size_t)B_ * NSLICES * TOPK * sizeof(float));

    dim3 g1(NSLICES, B_ / MROWS);
    knn_gemm_topk<<<g1, 512, 0, stream>>>(feat, anch, wsval, wsidx);
    knn_merge<<<B_, 32, 0, stream>>>(wsval, wsidx, labels, albl, out);
}